// GNNRecommender_28862180229821
// MI455X (gfx1250) — compile-verified
//
#include <hip/hip_runtime.h>
#include <hip/hip_bf16.h>

#define HID64 64

typedef float v2f __attribute__((ext_vector_type(2)));
typedef float v8f __attribute__((ext_vector_type(8)));

// ---------------------------------------------------------------- degrees ---
__global__ void gcn_deg_init(float* __restrict__ deg, int n) {
    int i = blockIdx.x * blockDim.x + threadIdx.x;
    if (i < n) deg[i] = 1.0f;                      // self-loop contribution
}

__global__ void gcn_deg_count(const int* __restrict__ dst, float* __restrict__ deg, int e) {
    int i = blockIdx.x * blockDim.x + threadIdx.x;
    if (i < e) atomicAdd(&deg[dst[i]], 1.0f);
}

__global__ void gcn_deg_rsqrt(float* __restrict__ deg, int n) {
    int i = blockIdx.x * blockDim.x + threadIdx.x;
    if (i < n) {
        float d = deg[i];
        deg[i] = (d > 0.0f) ? rsqrtf(d) : 0.0f;    // becomes dinv[]
    }
}

// ------------------------------------------------------- f32 WMMA GEMM -------
// C[n_rows, 64] = A[n_rows, K] @ W[64, K]^T     (W row-major, as in the ref)
//
// Block = 256 threads (8 waves). W is cooperatively staged into LDS in WMMA
// B-fragment order [step][colgroup][lane] (b64 each): every lane reads
// base + lane*8  -> all 64 LDS banks touched exactly once, conflict-free.
// Each wave computes a 32-row x 64-col tile: 8 accumulators, 8 WMMAs per
// K-step sharing 4 LDS B-fragments. A fragments are software-pipelined.
//
// V_WMMA_F32_16X16X4_F32 layouts (ISA 7.12.2):
//  A 16x4 : lanes 0-15 row M=l16 hold K={0,1}; lanes 16-31 hold K={2,3}
//  B 4x16 : vgpr0 = K=0 (lanes 0-15) / K=2 (lanes 16-31); vgpr1 = K=1 / K=3
//  D 16x16: vgpr r = M=r (lanes 0-15, N=lane) / M=r+8 (lanes 16-31, N=l16)
__global__ void __launch_bounds__(256)
gcn_gemm_wmma_f32(const float* __restrict__ A, const float* __restrict__ W,
                  float* __restrict__ C, int n_rows, int K) {
    __shared__ v2f ldsw[4096];                     // 32 KB, covers K up to 128

    const int tid  = threadIdx.x;
    const int lane = tid & 31;
    const int wave = tid >> 5;
    const int l16  = lane & 15;
    const int half = lane >> 4;

    // ---- stage W into LDS in fragment order: entry i = (step, j, lane) ----
    const int entries = 32 * K;                    // (K/4 steps) * 4 groups * 32 lanes
    for (int i = tid; i < entries; i += 256) {
        const int step = i >> 7;
        const int rem  = i & 127;
        const int fj   = rem >> 5;
        const int fl   = rem & 31;
        const int n    = 16 * fj + (fl & 15);
        const int k    = 4 * step + 2 * (fl >> 4);
        ldsw[i] = *(const v2f*)(W + (size_t)n * K + k);
    }
    __syncthreads();

    const int row0 = (blockIdx.x * 8 + wave) * 32;
    if (row0 + 32 > n_rows) return;                // wave-uniform: EXEC stays all-1s

    v8f acc0 = {}; v8f acc1 = {}; v8f acc2 = {}; v8f acc3 = {};
    v8f acc4 = {}; v8f acc5 = {}; v8f acc6 = {}; v8f acc7 = {};

    const float* arowA = A + (size_t)(row0 +      l16) * K + 2 * half;
    const float* arowB = A + (size_t)(row0 + 16 + l16) * K + 2 * half;

    const int nsteps = K >> 2;
    v2f aA = *(const v2f*)(arowA);
    v2f aB = *(const v2f*)(arowB);

#pragma unroll 2
    for (int s = 0; s < nsteps; ++s) {
        const int sn = (s + 1 < nsteps) ? s + 1 : s;
        v2f aAn = *(const v2f*)(arowA + 4 * sn);   // prefetch next A fragments
        v2f aBn = *(const v2f*)(arowB + 4 * sn);

        const v2f* bb = &ldsw[s * 128 + lane];
        v2f b0 = bb[ 0];
        v2f b1 = bb[32];
        v2f b2 = bb[64];
        v2f b3 = bb[96];

        acc0 = __builtin_amdgcn_wmma_f32_16x16x4_f32(false, aA, false, b0, (short)0, acc0, false, false);
        acc1 = __builtin_amdgcn_wmma_f32_16x16x4_f32(false, aA, false, b1, (short)0, acc1, false, false);
        acc2 = __builtin_amdgcn_wmma_f32_16x16x4_f32(false, aA, false, b2, (short)0, acc2, false, false);
        acc3 = __builtin_amdgcn_wmma_f32_16x16x4_f32(false, aA, false, b3, (short)0, acc3, false, false);
        acc4 = __builtin_amdgcn_wmma_f32_16x16x4_f32(false, aB, false, b0, (short)0, acc4, false, false);
        acc5 = __builtin_amdgcn_wmma_f32_16x16x4_f32(false, aB, false, b1, (short)0, acc5, false, false);
        acc6 = __builtin_amdgcn_wmma_f32_16x16x4_f32(false, aB, false, b2, (short)0, acc6, false, false);
        acc7 = __builtin_amdgcn_wmma_f32_16x16x4_f32(false, aB, false, b3, (short)0, acc7, false, false);

        aA = aAn;
        aB = aBn;
    }

    float* crow = C + (size_t)row0 * HID64;
#pragma unroll
    for (int r = 0; r < 8; ++r) {
        const int m = r + 8 * half;
        float* oA = crow + (size_t)m * HID64 + l16;
        float* oB = oA + 16 * HID64;
        oA[ 0] = acc0[r];  oA[16] = acc1[r];  oA[32] = acc2[r];  oA[48] = acc3[r];
        oB[ 0] = acc4[r];  oB[16] = acc5[r];  oB[32] = acc6[r];  oB[48] = acc7[r];
    }
}

// ---------------------------------------------------- agg init (fused) ------
// agg[n,:] = tmp[n,:]*dinv[n]^2 + bias    (self-loop term + bias, no atomics)
__global__ void gcn_agg_init(const float* __restrict__ tmp, const float* __restrict__ dinv,
                             const float* __restrict__ bias, float* __restrict__ agg,
                             long long n4) {
    long long i = (long long)blockIdx.x * blockDim.x + threadIdx.x;
    if (i >= n4) return;
    const int n = (int)(i >> 4);                   // 16 float4 per 64-wide row
    const int q = (int)(i & 15);
    const float di = dinv[n];
    const float w  = di * di;
    float4 t4 = ((const float4*)tmp)[i];
    float4 b4 = ((const float4*)bias)[q];
    float4 o;
    o.x = t4.x * w + b4.x;
    o.y = t4.y * w + b4.y;
    o.z = t4.z * w + b4.z;
    o.w = t4.w * w + b4.w;
    ((float4*)agg)[i] = o;
}

// ------------------------------------------------------------ edge scatter --
// 16 threads per edge; each handles 4 contiguous floats of the 64-wide row.
__global__ void gcn_edge_scatter(const int* __restrict__ src, const int* __restrict__ dst,
                                 const float* __restrict__ dinv,
                                 const float* __restrict__ h, float* __restrict__ agg,
                                 int n_edges) {
    long long t = (long long)blockIdx.x * blockDim.x + threadIdx.x;
    int e = (int)(t >> 4);
    if (e >= n_edges) return;
    int q = (int)(t & 15);
    int s = src[e], d = dst[e];
    float w = dinv[s] * dinv[d];
    const float4 v = *(const float4*)(h + (size_t)s * HID64 + q * 4);
    float* o = agg + (size_t)d * HID64 + q * 4;
    atomicAdd(o + 0, v.x * w);
    atomicAdd(o + 1, v.y * w);
    atomicAdd(o + 2, v.z * w);
    atomicAdd(o + 3, v.w * w);
}

// ------------------------------------------------------------------- relu ---
__global__ void gcn_relu4(float* __restrict__ p, long long n4) {
    long long i = (long long)blockIdx.x * blockDim.x + threadIdx.x;
    if (i >= n4) return;
    float4 v = ((const float4*)p)[i];
    v.x = v.x > 0.f ? v.x : 0.f;
    v.y = v.y > 0.f ? v.y : 0.f;
    v.z = v.z > 0.f ? v.z : 0.f;
    v.w = v.w > 0.f ? v.w : 0.f;
    ((float4*)p)[i] = v;
}

// ------------------------------------------------------------------- score --
__global__ void gcn_score(const float* __restrict__ h, const int* __restrict__ users,
                          const int* __restrict__ movies, const float* __restrict__ fcw,
                          const float* __restrict__ fcb, float* __restrict__ out, int batch) {
    int b = blockIdx.x * blockDim.x + threadIdx.x;
    if (b >= batch) return;
    const float4* hu = (const float4*)(h + (size_t)users[b]  * HID64);
    const float4* hm = (const float4*)(h + (size_t)movies[b] * HID64);
    const float4* fw = (const float4*)fcw;
    float s = 0.0f;
#pragma unroll
    for (int j = 0; j < HID64 / 4; ++j) {
        float4 u = hu[j], m = hm[j], w = fw[j];
        s += u.x * m.x * w.x + u.y * m.y * w.y + u.z * m.z * w.z + u.w * m.w * w.w;
    }
    out[b] = s + fcb[0];
}

// ------------------------------------------------------------------ launch --
extern "C" void kernel_launch(void* const* d_in, const int* in_sizes, int n_in,
                              void* d_out, int out_size, void* d_ws, size_t ws_size,
                              hipStream_t stream) {
    const float* x      = (const float*)d_in[0];
    const int*   edges  = (const int*)  d_in[1];
    const int*   users  = (const int*)  d_in[2];
    const int*   movies = (const int*)  d_in[3];
    const float* W1     = (const float*)d_in[4];
    const float* b1     = (const float*)d_in[5];
    const float* W2     = (const float*)d_in[6];
    const float* b2     = (const float*)d_in[7];
    const float* fcw    = (const float*)d_in[8];
    const float* fcb    = (const float*)d_in[9];
    float*       out    = (float*)d_out;

    const int HID = in_sizes[5];              // 64
    const int IN  = in_sizes[4] / HID;        // 128
    const int N   = in_sizes[0] / IN;         // 100000
    const int E   = in_sizes[1] / 2;          // 1200000
    const int B   = in_sizes[2];              // 8192

    const int* src = edges;
    const int* dst = edges + E;

    float* dinv = (float*)d_ws;                       // N
    float* bufT = dinv + N;                           // N*HID  (GEMM output)
    float* bufH = bufT + (size_t)N * HID;             // N*HID  (aggregated/relu)

    const int T = 256;
    const unsigned gN  = (unsigned)((N + T - 1) / T);
    const unsigned gE  = (unsigned)((E + T - 1) / T);
    const unsigned gG  = (unsigned)(((N + 31) / 32 + 7) / 8);       // 8 waves * 32 rows
    const long long nh4 = (long long)N * HID / 4;
    const unsigned g4  = (unsigned)((nh4 + T - 1) / T);
    const unsigned gS  = (unsigned)(((long long)E * 16 + T - 1) / T);
    const unsigned gB  = (unsigned)((B + T - 1) / T);

    // normalization: deg (incl. self-loop) -> dinv
    gcn_deg_init <<<gN, T, 0, stream>>>(dinv, N);
    gcn_deg_count<<<gE, T, 0, stream>>>(dst, dinv, E);
    gcn_deg_rsqrt<<<gN, T, 0, stream>>>(dinv, N);

    // ---- layer 1: tmp = x @ W1^T ; agg = self+bias ; scatter ; relu -> bufH
    gcn_gemm_wmma_f32<<<gG, T, 0, stream>>>(x, W1, bufT, N, IN);
    gcn_agg_init     <<<g4, T, 0, stream>>>(bufT, dinv, b1, bufH, nh4);
    gcn_edge_scatter <<<gS, T, 0, stream>>>(src, dst, dinv, bufT, bufH, E);
    gcn_relu4        <<<g4, T, 0, stream>>>(bufH, nh4);

    // ---- layer 2: tmp = h1 @ W2^T ; agg ; scatter ; relu -> bufH
    gcn_gemm_wmma_f32<<<gG, T, 0, stream>>>(bufH, W2, bufT, N, HID);
    gcn_agg_init     <<<g4, T, 0, stream>>>(bufT, dinv, b2, bufH, nh4);
    gcn_edge_scatter <<<gS, T, 0, stream>>>(src, dst, dinv, bufT, bufH, E);
    gcn_relu4        <<<g4, T, 0, stream>>>(bufH, nh4);

    // ---- pair scoring
    gcn_score<<<gB, T, 0, stream>>>(bufH, users, movies, fcw, fcb, out, B);
}